// AttentionMLP_36240934043956
// MI455X (gfx1250) — compile-verified
//
#include <hip/hip_runtime.h>
#include <hip/hip_bf16.h>

// Problem constants (from reference): B=2, T=512, C=128.
#define B_ 2
#define T_ 512
#define C_ 128
#define NEG_INF_ -1e9f
#define LPAD 132   // LDS row pad: 132 % 64-bank pattern -> conflict-free across 16 j-lanes

typedef __attribute__((ext_vector_type(2))) float v2f;
typedef __attribute__((ext_vector_type(8))) float v8f;

__device__ __forceinline__ float sigmoid_fast(float x) {
    // sigmoid(x) = 1/(1+exp(-x)); v_exp_f32 + v_rcp_f32 (trans pipe)
    float e = __expf(-x);
    return __builtin_amdgcn_rcpf(1.0f + e);
}

// ---------------------------------------------------------------------------
// Out[m,n] = sum_k X[m,k]*W[n,k] + b1[n] (+ b2[n])   (torch Linear: x @ W^T + b)
// One wave per 16x16 tile, fp32 WMMA, K in steps of 4.
// ---------------------------------------------------------------------------
__global__ __launch_bounds__(32) void gemm_xwT_bias_kernel(
    const float* __restrict__ X, const float* __restrict__ W,
    const float* __restrict__ b1, const float* __restrict__ b2,
    float* __restrict__ Out, int M, int N, int K)
{
    const int lane = threadIdx.x;
    const int half = lane >> 4;   // 0: lanes 0-15 (K=0,1 / rows 0-7), 1: lanes 16-31 (K=2,3 / rows 8-15)
    const int l    = lane & 15;
    const int row0 = blockIdx.x * 16;
    const int col0 = blockIdx.y * 16;

    const float* xrow = X + (size_t)(row0 + l) * K;   // A: M = l across lanes
    const float* wrow = W + (size_t)(col0 + l) * K;   // B(=W^T chunk): N = l across lanes

    v8f acc = {};
    for (int k0 = 0; k0 < K; k0 += 4) {
        const int ka = k0 + 2 * half;
        v2f a;  a.x  = xrow[ka]; a.y  = xrow[ka + 1];   // contiguous pair -> b64 load
        v2f bm; bm.x = wrow[ka]; bm.y = wrow[ka + 1];
        acc = __builtin_amdgcn_wmma_f32_16x16x4_f32(
            false, a, false, bm, (short)0, acc, false, false);
    }

    float bc = b1[col0 + l];
    if (b2) bc += b2[col0 + l];
#pragma unroll
    for (int r = 0; r < 8; ++r) {
        const int m = r + 8 * half;                     // C/D layout: VGPR r -> row r / r+8
        Out[(size_t)(row0 + m) * N + col0 + l] = acc[r] + bc;
    }
}

// ---------------------------------------------------------------------------
// Out[m,n] = sum_k A[m,k]*Bm[k,n]  (row-major x row-major), batched via blockIdx.z
// Used for attn(TxT) @ Kp(TxC).
// ---------------------------------------------------------------------------
__global__ __launch_bounds__(32) void gemm_ab_kernel(
    const float* __restrict__ A, const float* __restrict__ Bm,
    float* __restrict__ Out, int M, int N, int K,
    size_t strideA, size_t strideB, size_t strideC)
{
    const int lane = threadIdx.x;
    const int half = lane >> 4;
    const int l    = lane & 15;
    const int row0 = blockIdx.x * 16;
    const int col0 = blockIdx.y * 16;

    const float* a_ = A  + (size_t)blockIdx.z * strideA;
    const float* b_ = Bm + (size_t)blockIdx.z * strideB;
    float*       o_ = Out + (size_t)blockIdx.z * strideC;

    const float* arow = a_ + (size_t)(row0 + l) * K;

    v8f acc = {};
    for (int k0 = 0; k0 < K; k0 += 4) {
        const int ka = k0 + 2 * half;
        v2f a;  a.x  = arow[ka]; a.y = arow[ka + 1];
        v2f bm; bm.x = b_[(size_t)ka * N + col0 + l];
                bm.y = b_[(size_t)(ka + 1) * N + col0 + l];
        acc = __builtin_amdgcn_wmma_f32_16x16x4_f32(
            false, a, false, bm, (short)0, acc, false, false);
    }
#pragma unroll
    for (int r = 0; r < 8; ++r) {
        const int m = r + 8 * half;
        o_[(size_t)(row0 + m) * N + col0 + l] = acc[r];
    }
}

// ---------------------------------------------------------------------------
// logits[b,i,j] = sum_c sigmoid(Qb[b,i,c] + Kp[b,j,c]); masked write.
// 16x16 (i,j) tile per 256-thread block; Q/K tiles staged in padded LDS.
// Dominant kernel: ~67M sigmoid evals -> trans-pipe bound, stays in L2.
// ---------------------------------------------------------------------------
__global__ __launch_bounds__(256) void logits_kernel(
    const float* __restrict__ Qb, const float* __restrict__ Kp,
    const unsigned char* __restrict__ mask, float* __restrict__ logits)
{
    __shared__ __align__(16) float sq[16 * LPAD];
    __shared__ __align__(16) float sk[16 * LPAD];

    const int b  = blockIdx.z;
    const int i0 = blockIdx.y * 16;
    const int j0 = blockIdx.x * 16;
    const int t  = threadIdx.x;

    const float* Qbase = Qb + ((size_t)b * T_ + i0) * C_;
    const float* Kbase = Kp + ((size_t)b * T_ + j0) * C_;

    // Stage 16x128 tiles as float4 (512 vec4 per tile; 2 per thread)
    for (int v = t; v < 16 * (C_ / 4); v += 256) {
        const int r  = v >> 5;        // C_/4 == 32 vec4 per row
        const int c4 = v & 31;
        const float4 qv = ((const float4*)(Qbase + (size_t)r * C_))[c4];
        const float4 kv = ((const float4*)(Kbase + (size_t)r * C_))[c4];
        *(float4*)&sq[r * LPAD + c4 * 4] = qv;
        *(float4*)&sk[r * LPAD + c4 * 4] = kv;
    }
    __syncthreads();

    const int i = t >> 4;
    const int j = t & 15;
    const float* qr = &sq[i * LPAD];
    const float* kr = &sk[j * LPAD];

    float acc = 0.0f;
#pragma unroll 8
    for (int c = 0; c < C_; c += 4) {
        const float4 qv = *(const float4*)&qr[c];
        const float4 kv = *(const float4*)&kr[c];
        acc += sigmoid_fast(qv.x + kv.x);
        acc += sigmoid_fast(qv.y + kv.y);
        acc += sigmoid_fast(qv.z + kv.z);
        acc += sigmoid_fast(qv.w + kv.w);
    }

    const size_t idx = ((size_t)b * T_ + (i0 + i)) * T_ + (j0 + j);
    logits[idx] = mask[idx] ? acc : NEG_INF_;   // numpy/jax bool == 1 byte
}

// ---------------------------------------------------------------------------
// Row softmax over T_=512: one wave per row, 16 elems/lane in registers,
// wave32 shuffle reductions for max and sum.
// ---------------------------------------------------------------------------
__global__ __launch_bounds__(32) void softmax_kernel(
    const float* __restrict__ logits, float* __restrict__ attn)
{
    const int row  = blockIdx.x;     // 0 .. B*T-1
    const int lane = threadIdx.x;
    const float* lr = logits + (size_t)row * T_;

    float v[T_ / 32];
    float m = -3.4e38f;
#pragma unroll
    for (int u = 0; u < T_ / 32; ++u) {
        v[u] = lr[lane + u * 32];
        m = fmaxf(m, v[u]);
    }
#pragma unroll
    for (int off = 16; off > 0; off >>= 1)
        m = fmaxf(m, __shfl_xor(m, off, 32));

    float s = 0.0f;
#pragma unroll
    for (int u = 0; u < T_ / 32; ++u) {
        v[u] = __expf(v[u] - m);
        s += v[u];
    }
#pragma unroll
    for (int off = 16; off > 0; off >>= 1)
        s += __shfl_xor(s, off, 32);

    const float inv = __builtin_amdgcn_rcpf(s);
    float* ar = attn + (size_t)row * T_;
#pragma unroll
    for (int u = 0; u < T_ / 32; ++u)
        ar[lane + u * 32] = v[u] * inv;
}

// ---------------------------------------------------------------------------
extern "C" void kernel_launch(void* const* d_in, const int* in_sizes, int n_in,
                              void* d_out, int out_size, void* d_ws, size_t ws_size,
                              hipStream_t stream)
{
    const float*         q    = (const float*)d_in[0];
    const float*         k    = (const float*)d_in[1];
    const unsigned char* mask = (const unsigned char*)d_in[2];  // bool tensor, 1 B/elem
    const float*         Wq_w = (const float*)d_in[3];
    const float*         Wq_b = (const float*)d_in[4];
    const float*         Wk_w = (const float*)d_in[5];
    const float*         Wk_b = (const float*)d_in[6];
    const float*         bias = (const float*)d_in[7];
    const float*         Wv_w = (const float*)d_in[8];
    const float*         Wv_b = (const float*)d_in[9];

    float* out    = (float*)d_out;                         // [B*T*C] final output
    float* logits = out + (size_t)B_ * T_ * C_;            // [B*T*T] attention_logits

    // Workspace: 3 * B*T*C + B*T*T floats = 3.5 MB
    float* Qb     = (float*)d_ws;                          // Q proj + Wq_b + bias folded in
    float* Kp     = Qb + (size_t)B_ * T_ * C_;             // K proj + Wk_b
    float* attn   = Kp + (size_t)B_ * T_ * C_;             // softmax(logits)
    float* outTmp = attn + (size_t)B_ * T_ * T_;           // attn @ Kp

    const dim3 gProj(B_ * T_ / 16, C_ / 16);               // 64 x 8 tiles
    gemm_xwT_bias_kernel<<<gProj, 32, 0, stream>>>(q, Wq_w, Wq_b, bias, Qb,
                                                   B_ * T_, C_, C_);
    gemm_xwT_bias_kernel<<<gProj, 32, 0, stream>>>(k, Wk_w, Wk_b, nullptr, Kp,
                                                   B_ * T_, C_, C_);

    const dim3 gLog(T_ / 16, T_ / 16, B_);                 // 32 x 32 x 2 blocks
    logits_kernel<<<gLog, 256, 0, stream>>>(Qb, Kp, mask, logits);

    softmax_kernel<<<B_ * T_, 32, 0, stream>>>(logits, attn);

    const dim3 gAV(T_ / 16, C_ / 16, B_);                  // 32 x 8 x 2 tiles
    gemm_ab_kernel<<<gAV, 32, 0, stream>>>(attn, Kp, outTmp, T_, C_, T_,
                                           (size_t)T_ * T_, (size_t)T_ * C_,
                                           (size_t)T_ * C_);

    gemm_xwT_bias_kernel<<<gProj, 32, 0, stream>>>(outTmp, Wv_w, Wv_b, nullptr, out,
                                                   B_ * T_, C_, C_);
}